// DecodePredictions_12197707120949
// MI455X (gfx1250) — compile-verified
//
#include <hip/hip_runtime.h>

// YOLO decode for MI455X (gfx1250).
//
// Roofline: output = B*8400*80*6 floats = 258 MB written, inputs ~46 MB read
// => ~13 us floor at 23.3 TB/s. Pure elementwise + rank-1 broadcast: no
// matrix contraction, so WMMA does not apply. The gfx1250-specific wins are:
//   * async global->LDS staging (global_load_async_to_lds_b32 / ASYNCcnt /
//     s_wait_asynccnt). Fast path exploits that INST_OFFSET is added to BOTH
//     the LDS and global addresses: a contiguous 32-row region is staged with
//     9 async ops that differ only in their offset: immediate (GVS form,
//     uniform SGPR base).
//   * 12 aligned b128 non-temporal stores per thread (output >> 192 MB L2,
//     stream it past the caches).
//   * 8 classes per thread so per-anchor decode math is amortized; all int
//     divisions use compile-time-constant divisors (magic multiplies).

#define CLS   80
#define ROW   85                  // 5 + CLS floats per anchor row
#define API   8400                // anchors per image: 80^2 + 40^2 + 20^2
#define G     32                  // anchors per block
#define CPT   8                   // classes per thread
#define TPA   (CLS / CPT)         // 10 threads per anchor
#define TPB   (G * TPA)           // 320 threads (10 waves of wave32)
#define STAGE (G * ROW)           // 2720 floats staged per block

typedef float v4f __attribute__((ext_vector_type(4)));

__device__ __forceinline__ float sigm(float x) {
  return 1.0f / (1.0f + __expf(-x));
}

__global__ __launch_bounds__(TPB) void yolo_decode_kernel(
    const float* __restrict__ p0,   // [B, 80, 80, 85]
    const float* __restrict__ p1,   // [B, 40, 40, 85]
    const float* __restrict__ p2,   // [B, 20, 20, 85]
    float* __restrict__ out,        // [B, 8400*80, 6]
    int total) {
  __shared__ float sm[STAGE];
  const int t   = threadIdx.x;
  const int ga0 = blockIdx.x * G;

  // ---- Stage G rows (85 floats each) into LDS with CDNA5 async copies ----
  const int gaL = ga0 + G - 1;
  const int bF = ga0 / API, aF = ga0 - bF * API;
  const int bL = gaL / API, aL = gaL - bL * API;
  const int lF = aF < 6400 ? 0 : (aF < 8000 ? 1 : 2);
  const int lL = aL < 6400 ? 0 : (aL < 8000 ? 1 : 2);

  if (bF == bL && lF == lL && gaL < total) {
    // Fast path: the 32 rows are contiguous in one (batch, level) region.
    const float* base;
    if (lF == 0)      base = p0 + ((size_t)bF * 6400 +  aF        ) * ROW;
    else if (lF == 1) base = p1 + ((size_t)bF * 1600 + (aF - 6400)) * ROW;
    else              base = p2 + ((size_t)bF *  400 + (aF - 8000)) * ROW;
    // Same VGPR operands for all 9 copies; only the immediate offset steps.
    // (the immediate offset is added to both the LDS and the global address)
    unsigned lds0 = (unsigned)(size_t)(&sm[0]) + (unsigned)(t * 4);
    unsigned goff = (unsigned)(t * 4);
    asm volatile(
        "global_load_async_to_lds_b32 %0, %1, %2 offset:0\n\t"
        "global_load_async_to_lds_b32 %0, %1, %2 offset:1280\n\t"
        "global_load_async_to_lds_b32 %0, %1, %2 offset:2560\n\t"
        "global_load_async_to_lds_b32 %0, %1, %2 offset:3840\n\t"
        "global_load_async_to_lds_b32 %0, %1, %2 offset:5120\n\t"
        "global_load_async_to_lds_b32 %0, %1, %2 offset:6400\n\t"
        "global_load_async_to_lds_b32 %0, %1, %2 offset:7680\n\t"
        "global_load_async_to_lds_b32 %0, %1, %2 offset:8960"
        :: "v"(lds0), "v"(goff), "s"(base) : "memory");
    if (t < STAGE - 8 * TPB)   // tail: 160 slots
      asm volatile(
          "global_load_async_to_lds_b32 %0, %1, %2 offset:10240"
          :: "v"(lds0), "v"(goff), "s"(base) : "memory");
  } else {
    // Slow path (block straddles a level/batch boundary; ~1% of blocks).
    for (int s = t; s < STAGE; s += TPB) {
      const int i = s / ROW, e = s - i * ROW;
      const int ga = ga0 + i;
      if (ga < total) {
        const int b = ga / API, a = ga - b * API;
        const float* src;
        if (a < 6400)      src = p0 + ((size_t)b * 6400 +  a        ) * ROW + e;
        else if (a < 8000) src = p1 + ((size_t)b * 1600 + (a - 6400)) * ROW + e;
        else               src = p2 + ((size_t)b *  400 + (a - 8000)) * ROW + e;
        unsigned           lds_off = (unsigned)(size_t)(&sm[s]);
        unsigned long long gaddr   = (unsigned long long)(size_t)src;
        asm volatile("global_load_async_to_lds_b32 %0, %1, off"
                     :: "v"(lds_off), "v"(gaddr) : "memory");
      }
    }
  }
  asm volatile("s_wait_asynccnt 0x0" ::: "memory");
  __syncthreads();

  // ---- Decode: thread t = (local anchor i = t/10, class group q = t%10) ----
  const int i  = t / TPA;
  const int q  = t - i * TPA;
  const int ga = ga0 + i;
  if (ga >= total) return;

  const int b = ga / API;
  const int a = ga - b * API;
  int ix, iy; float stride;
  if (a < 6400)      { iy = a / 80;              ix = a - iy * 80;          stride = 8.0f;  }
  else if (a < 8000) { const int n = a - 6400; iy = n / 40; ix = n - iy * 40; stride = 16.0f; }
  else               { const int n = a - 8000; iy = n / 20; ix = n - iy * 20; stride = 32.0f; }

  const float* r = &sm[i * ROW];
  const float px = r[0], py = r[1], pw = r[2], ph = r[3];
  const float obj = sigm(r[4]);

  // Faithful to reference: (p + g) * stride / 640, then * 640 on output.
  const float xr = (px + (float)ix) * stride / 640.0f;
  const float yr = (py + (float)iy) * stride / 640.0f;
  const float wr = __expf(pw) * stride / 640.0f;
  const float hr = __expf(ph) * stride / 640.0f;
  const float x1 = xr * 640.0f;
  const float y1 = yr * 640.0f;
  const float x2 = (xr + wr) * 640.0f;
  const float y2 = (yr + hr) * 640.0f;

  // 8 classes -> 48 floats -> 12 aligned b128 streaming stores.
  float* dst = out + (size_t)ga * (CLS * 6) + (size_t)(q * CPT) * 6;
#pragma unroll
  for (int k = 0; k < CPT; k += 2) {
    const int   c  = q * CPT + k;
    const float s0 = obj * sigm(r[5 + c]);
    const float s1 = obj * sigm(r[6 + c]);
    v4f A  = { x1, y1, x2, y2 };
    v4f B4 = { (float)c, s0, x1, y1 };
    v4f C4 = { x2, y2, (float)(c + 1), s1 };
    __builtin_nontemporal_store(A,  (v4f*)(dst + 6 * k + 0));
    __builtin_nontemporal_store(B4, (v4f*)(dst + 6 * k + 4));
    __builtin_nontemporal_store(C4, (v4f*)(dst + 6 * k + 8));
  }
}

extern "C" void kernel_launch(void* const* d_in, const int* in_sizes, int n_in,
                              void* d_out, int out_size, void* d_ws, size_t ws_size,
                              hipStream_t stream) {
  // d_in: [0]=images (shape-only, never read), [1]=pred0, [2]=pred1, [3]=pred2
  const float* p0  = (const float*)d_in[1];
  const float* p1  = (const float*)d_in[2];
  const float* p2  = (const float*)d_in[3];
  float*       out = (float*)d_out;

  const int B      = in_sizes[1] / (80 * 80 * (5 + CLS));
  const int total  = B * API;
  const int blocks = (total + G - 1) / G;   // 4200 for B=16

  yolo_decode_kernel<<<blocks, TPB, 0, stream>>>(p0, p1, p2, out, total);
}